// TransitionProbLayerBeta_43430709297765
// MI455X (gfx1250) — compile-verified
//
#include <hip/hip_runtime.h>

typedef __attribute__((ext_vector_type(2))) float v2f;
typedef __attribute__((ext_vector_type(8))) float v8f;

#define DAYS 513
#define DUR  512
#define BATCH 8192
#define BASIS 1e-10f

__device__ __forceinline__ float bperm(float v, int srcLane) {
    // ds_bpermute_b32: dst[lane] = src[index[lane]/4 & 31] (wave32 uses bits [6:2])
    return __int_as_float(__builtin_amdgcn_ds_bpermute(srcLane << 2, __float_as_int(v)));
}

__global__ __launch_bounds__(256) void TransitionProbQ_kernel(
    const float* __restrict__ adstock,   // [8192, 3, 513]
    const float* __restrict__ mu,        // [3]
    const float* __restrict__ beta,      // [9]  (3x3 row-major: beta[c*3 + k])
    float* __restrict__ out)             // [8192, 512, 3, 3]
{
    __shared__ __align__(16) float smem[8 * 288];   // 8 waves * 32 rows * 9 floats

    const int tid  = threadIdx.x;
    const int lane = tid & 31;
    const int wv   = tid >> 5;
    const int half = lane >> 4;        // 0: lanes 0-15 (K=0,1), 1: lanes 16-31 (K=2,3)
    const int m    = lane & 15;

    // This lane's row of the (b,d) grid. Grid is exact: 16384 blocks * 256 = 4,194,304 rows.
    const int r = blockIdx.x * 256 + tid;
    const int b = r >> 9;              // dur = 512
    const int d = r & 511;
    const int abase = b * (3 * DAYS) + (d + 1);   // adstock[:, :, 1:]

    const float a0 = adstock[abase];
    const float a1 = adstock[abase + DAYS];
    const float a2 = adstock[abase + 2 * DAYS];

    // ---- A operand (16x4 f32): A[M=k][K=c] = beta_r[c][k]; row 0 and rows>=4 are zero.
    // Layout: lanes 0-15: v0=K0,v1=K1 for M=lane; lanes 16-31: v0=K2,v1=K3 for M=lane-16.
    const bool kv = (m >= 1) && (m <= 3);
    v2f A;
    A.x = kv ? beta[(half ? 6 : 0) + (m - 1)] : 0.0f;   // c = half?2:0
    A.y = (kv && !half) ? beta[3 + (m - 1)] : 0.0f;     // c=1 (K=3 is zero pad)

    // ---- B operands (4x16 f32): B[K=c][N=n] = adstock(row tileBase+n, channel c)
    // Same K-split layout as A. Cross-lane moves (must run with full EXEC):
    const float p_a2 = bperm(a2, m);                 // lane L: a2 from lane L&15
    const float p_a0 = bperm(a0, (lane + 16) & 31);  // lane L: a0 from lane L+16
    const float p_a1 = bperm(a1, (lane + 16) & 31);

    v2f B1, B2;
    B1.x = half ? p_a2 : a0;   // tile rows base..base+15
    B1.y = half ? 0.0f : a1;
    B2.x = half ? a2   : p_a0; // tile rows base+16..base+31
    B2.y = half ? 0.0f : p_a1;

    v8f cz = {};
    // D[M=k][N=row]: lanes 0-15 hold rows (N=lane) with O_k in accumulator element k.
    v8f d1v = __builtin_amdgcn_wmma_f32_16x16x4_f32(false, A, false, B1, (short)0, cz, false, false);
    v8f d2v = __builtin_amdgcn_wmma_f32_16x16x4_f32(false, A, false, B2, (short)0, cz, false, false);

    // Redistribute: lane L>=16 takes tile-2 results from lane L-16.
    const float t1 = bperm(d2v[1], m);
    const float t2 = bperm(d2v[2], m);
    const float t3 = bperm(d2v[3], m);
    const float o1 = half ? t1 : d1v[1];
    const float o2 = half ? t2 : d1v[2];
    const float o3 = half ? t3 : d1v[3];

    // ---- scalar tail per row
    const float e01 = expf(mu[0] + o1);
    const float e10 = expf(mu[1] + o2);
    const float e11 = expf(mu[2] + o3);
    const float den0 = 1.0f + e01;
    const float den1 = 1.0f + e10 + e11;

    float q[9];
    q[0] = fmaxf(1.0f / den0, BASIS);
    q[1] = fmaxf(e01 / den0, BASIS);
    q[2] = BASIS;
    q[3] = fmaxf(e10 / den1, BASIS);
    q[4] = fmaxf(1.0f / den1, BASIS);
    q[5] = fmaxf(e11 / den1, BASIS);
    q[6] = BASIS;
    q[7] = BASIS;
    q[8] = 1.0f;

    // ---- stage 36B/row in LDS (dword index 9L+i mod 64: conflict-free permutation)
    float* ws = &smem[wv * 288];
    #pragma unroll
    for (int i = 0; i < 9; ++i) ws[lane * 9 + i] = q[i];
    __syncthreads();

    // ---- coalesced writeback of the wave's 1152B block (16B-aligned: 32*36 = 1152)
    const size_t outBase = ((size_t)blockIdx.x * 256 + (size_t)wv * 32) * 9;
    const float4* s4 = (const float4*)ws;
    float4* o4 = (float4*)(out + outBase);
    o4[lane]      = s4[lane];        // dwords   0..127
    o4[32 + lane] = s4[32 + lane];   // dwords 128..255
    out[outBase + 256 + lane] = ws[256 + lane];  // dwords 256..287
}

__global__ void Sigmoid_tail_kernel(const float* __restrict__ ip,
                                    const float* __restrict__ cp,
                                    float* __restrict__ out) {
    out[0] = 1.0f / (1.0f + expf(-ip[0]));
    out[1] = 1.0f / (1.0f + expf(-cp[0]));
}

extern "C" void kernel_launch(void* const* d_in, const int* in_sizes, int n_in,
                              void* d_out, int out_size, void* d_ws, size_t ws_size,
                              hipStream_t stream) {
    const float* adstock = (const float*)d_in[0];
    const float* mu      = (const float*)d_in[1];
    const float* beta    = (const float*)d_in[2];
    const float* ip      = (const float*)d_in[3];
    const float* cp      = (const float*)d_in[4];
    float* out = (float*)d_out;

    // 8192*512 rows, 1 row/thread, 256 threads/block -> 16384 blocks (exact fit)
    TransitionProbQ_kernel<<<dim3(16384), dim3(256), 0, stream>>>(adstock, mu, beta, out);

    const size_t qElems = (size_t)BATCH * DUR * 9;   // 37,748,736
    Sigmoid_tail_kernel<<<dim3(1), dim3(1), 0, stream>>>(ip, cp, out + qElems);
}